// GAT_1580547975275
// MI455X (gfx1250) — compile-verified
//
#include <hip/hip_runtime.h>
#include <hip/hip_bf16.h>

#define NODES 100000
#define EDGES 1600000
#define F_INN 4
#define HID 8
#define HEADS 8
#define HF (HEADS * HID)   /* 64 */
#define NEG_SLOPE 0.2f

#define ROWS_MAIN ((NODES / 128) * 128)   /* 99968 rows via WMMA, no guards */

typedef __attribute__((ext_vector_type(2))) float v2f;
typedef __attribute__((ext_vector_type(8))) float v8f;

__device__ __forceinline__ float lrelu(float v) { return v > 0.f ? v : NEG_SLOPE * v; }

__device__ __forceinline__ void edge_fetch(const int* __restrict__ ei, int e, int& src, int& dst) {
  if (e < EDGES) { src = ei[e]; dst = ei[EDGES + e]; }
  else           { src = e - EDGES; dst = src; }
}

// load 8 consecutive, 32B-aligned floats as two b128s
__device__ __forceinline__ void load8(const float* __restrict__ p, float* __restrict__ v) {
  const float4* p4 = (const float4*)p;
  float4 a = p4[0], b = p4[1];
  v[0] = a.x; v[1] = a.y; v[2] = a.z; v[3] = a.w;
  v[4] = b.x; v[5] = b.y; v[6] = b.z; v[7] = b.w;
}

// ---------------------------------------------------------------------------
// Kernel 1a: h1 = x @ W1 for the first 99968 rows via V_WMMA_F32_16X16X4_F32.
// K=4 exactly matches F_IN. One wave = 16-node x 64-col tile = 4 WMMAs.
// A layout (16x4 f32): lanes 0-15 -> K={0,1}, lanes 16-31 -> K={2,3}.
// B layout (4x16 f32): VGPR v, lanes 0-15 -> K=v, lanes 16-31 -> K=2+v.
// C/D layout: VGPR r, lanes 0-15 -> M=r, lanes 16-31 -> M=8+r; lane%16 = N.
// Grid covers only full tiles: zero bounds checks, straight-line stores with
// immediate offsets (r*256B) off one base address.
// ---------------------------------------------------------------------------
__global__ __launch_bounds__(256) void k_gemm1(const float* __restrict__ x,
                                               const float* __restrict__ W1,
                                               float* __restrict__ h1) {
  const int wave = threadIdx.x >> 5;
  const int lane = threadIdx.x & 31;
  const int half = lane & 15;
  const bool hi  = lane >= 16;
  const int rowBase = blockIdx.x * 128 + wave * 16;

  const float* xp = x + (size_t)(rowBase + half) * F_INN + (hi ? 2 : 0);
  v2f a;
  a.x = xp[0];
  a.y = xp[1];

#pragma unroll
  for (int t = 0; t < 4; ++t) {
    const int col = t * 16 + half;
    v2f b;
    b.x = W1[(hi ? 2 : 0) * HF + col];
    b.y = W1[(hi ? 3 : 1) * HF + col];
    v8f c = {0.f, 0.f, 0.f, 0.f, 0.f, 0.f, 0.f, 0.f};
    c = __builtin_amdgcn_wmma_f32_16x16x4_f32(false, a, false, b, (short)0, c, false, false);
    float* hp = h1 + (size_t)(rowBase + (hi ? 8 : 0)) * HF + col;
#pragma unroll
    for (int r = 0; r < 8; ++r) hp[(size_t)r * HF] = c[r];   // 64B-contiguous per 16 lanes
  }
}

// ---------------------------------------------------------------------------
// Kernel 1b: tail rows [99968, 100000): 32x64 outputs, K=4 dot each (VALU).
// ---------------------------------------------------------------------------
__global__ void k_gemm1_tail(const float* __restrict__ x,
                             const float* __restrict__ W1,
                             float* __restrict__ h1) {
  const int idx = blockIdx.x * blockDim.x + threadIdx.x;
  const int row = ROWS_MAIN + (idx >> 6);
  const int col = idx & 63;
  if (row >= NODES) return;
  const float* xp = x + (size_t)row * F_INN;
  float s = 0.f;
#pragma unroll
  for (int k = 0; k < F_INN; ++k) s += xp[k] * W1[k * HF + col];
  h1[(size_t)row * HF + col] = s;
}

// ---------------------------------------------------------------------------
// Kernel 2: per-(node,head) attention scalars a_s = <h,att_src>, a_d = <h,att_dst>
// ---------------------------------------------------------------------------
__global__ void k_att1(const float* __restrict__ h1,
                       const float* __restrict__ att_s, const float* __restrict__ att_d,
                       float* __restrict__ a_s, float* __restrict__ a_d) {
  const int idx = blockIdx.x * blockDim.x + threadIdx.x;
  if (idx >= NODES * HEADS) return;
  const int node = idx >> 3;
  const int head = idx & 7;
  float hv[8], as[8], ad[8];
  load8(h1 + (size_t)node * HF + head * HID, hv);
  load8(att_s + head * HID, as);
  load8(att_d + head * HID, ad);
  float ss = 0.f, sd = 0.f;
#pragma unroll
  for (int c = 0; c < HID; ++c) { ss += hv[c] * as[c]; sd += hv[c] * ad[c]; }
  a_s[idx] = ss;
  a_d[idx] = sd;
}

// ---------------------------------------------------------------------------
// Init segment-softmax state for both layers
// ---------------------------------------------------------------------------
__global__ void k_init(float* __restrict__ m1, float* __restrict__ s1, float* __restrict__ acc1,
                       float* __restrict__ m2, float* __restrict__ s2, float* __restrict__ acc2) {
  const int idx = blockIdx.x * blockDim.x + threadIdx.x;
  if (idx < NODES * HF) acc1[idx] = 0.f;
  if (idx < NODES * HEADS) { m1[idx] = -3.0e38f; s1[idx] = 0.f; }
  if (idx < NODES) { m2[idx] = -3.0e38f; s2[idx] = 0.f; acc2[idx] = 0.f; }
}

// ---------------------------------------------------------------------------
// Layer-1 edge pass A: segment max over destinations (GLOBAL_ATOMIC_MAX_NUM_F32)
// ---------------------------------------------------------------------------
__global__ void k_max1(const int* __restrict__ ei,
                       const float* __restrict__ a_s, const float* __restrict__ a_d,
                       float* __restrict__ m1) {
  const int e = blockIdx.x * blockDim.x + threadIdx.x;
  if (e >= EDGES + NODES) return;
  __builtin_prefetch(ei + e + 4096, 0, 0);            // global_prefetch_b8, streamed edge list
  __builtin_prefetch(ei + EDGES + e + 4096, 0, 0);
  int src, dst;
  edge_fetch(ei, e, src, dst);
  float as[8], ad[8];
  load8(a_s + (size_t)src * HEADS, as);
  load8(a_d + (size_t)dst * HEADS, ad);
  float* mp = m1 + (size_t)dst * HEADS;
#pragma unroll
  for (int h = 0; h < HEADS; ++h) {
    atomicMax(&mp[h], lrelu(as[h] + ad[h]));
  }
}

// ---------------------------------------------------------------------------
// Layer-1 edge pass B: segment sum of exp(e - max)  (GLOBAL_ATOMIC_ADD_F32)
// ---------------------------------------------------------------------------
__global__ void k_sum1(const int* __restrict__ ei,
                       const float* __restrict__ a_s, const float* __restrict__ a_d,
                       const float* __restrict__ m1, float* __restrict__ s1) {
  const int e = blockIdx.x * blockDim.x + threadIdx.x;
  if (e >= EDGES + NODES) return;
  int src, dst;
  edge_fetch(ei, e, src, dst);
  float as[8], ad[8], mv[8];
  load8(a_s + (size_t)src * HEADS, as);
  load8(a_d + (size_t)dst * HEADS, ad);
  load8(m1 + (size_t)dst * HEADS, mv);
  float* sp = s1 + (size_t)dst * HEADS;
#pragma unroll
  for (int h = 0; h < HEADS; ++h) {
    atomicAdd(&sp[h], __expf(lrelu(as[h] + ad[h]) - mv[h]));
  }
}

// ---------------------------------------------------------------------------
// Layer-1 edge pass C: alpha-weighted message scatter-add (one (edge,head)/thread)
// ---------------------------------------------------------------------------
__global__ void k_agg1(const int* __restrict__ ei,
                       const float* __restrict__ a_s, const float* __restrict__ a_d,
                       const float* __restrict__ m1, const float* __restrict__ s1,
                       const float* __restrict__ h1, float* __restrict__ acc1) {
  const int idx = blockIdx.x * blockDim.x + threadIdx.x;
  if (idx >= (EDGES + NODES) * HEADS) return;
  const int e = idx >> 3;
  const int h = idx & 7;
  int src, dst;
  edge_fetch(ei, e, src, dst);
  const float v = lrelu(a_s[(size_t)src * HEADS + h] + a_d[(size_t)dst * HEADS + h]);
  const float alpha = __expf(v - m1[(size_t)dst * HEADS + h]) /
                      (s1[(size_t)dst * HEADS + h] + 1e-16f);
  float hs[8];
  load8(h1 + (size_t)src * HF + h * HID, hs);
  float* ap = acc1 + (size_t)dst * HF + h * HID;
#pragma unroll
  for (int c = 0; c < HID; ++c) {
    atomicAdd(&ap[c], hs[c] * alpha);
  }
}

// ---------------------------------------------------------------------------
// Layer-1 finalize fused with layer-2 transform:
// h1e = elu(acc1 + b1);  h2 = h1e @ W2;  as2/ad2 = h2 * att2 scalars
// (h1e itself is never needed again, only the fused scalar h2.)
// ---------------------------------------------------------------------------
__global__ void k_fin1(const float* __restrict__ acc1, const float* __restrict__ b1,
                       const float* __restrict__ W2,
                       const float* __restrict__ att_s2, const float* __restrict__ att_d2,
                       float* __restrict__ h2, float* __restrict__ as2, float* __restrict__ ad2) {
  const int n = blockIdx.x * blockDim.x + threadIdx.x;
  if (n >= NODES) return;
  const float* ap = acc1 + (size_t)n * HF;
  float s = 0.f;
#pragma unroll
  for (int j = 0; j < HF; ++j) {
    float v = ap[j] + b1[j];
    v = (v > 0.f) ? v : (__expf(v) - 1.f);   // ELU
    s += v * W2[j];
  }
  h2[n]  = s;
  as2[n] = s * att_s2[0];
  ad2[n] = s * att_d2[0];
}

// ---------------------------------------------------------------------------
// Layer-2 edge passes (single head)
// ---------------------------------------------------------------------------
__global__ void k_max2(const int* __restrict__ ei, const float* __restrict__ as2,
                       const float* __restrict__ ad2, float* __restrict__ m2) {
  const int e = blockIdx.x * blockDim.x + threadIdx.x;
  if (e >= EDGES + NODES) return;
  __builtin_prefetch(ei + e + 4096, 0, 0);
  __builtin_prefetch(ei + EDGES + e + 4096, 0, 0);
  int src, dst;
  edge_fetch(ei, e, src, dst);
  atomicMax(&m2[dst], lrelu(as2[src] + ad2[dst]));
}

__global__ void k_sum2(const int* __restrict__ ei, const float* __restrict__ as2,
                       const float* __restrict__ ad2, const float* __restrict__ m2,
                       float* __restrict__ s2) {
  const int e = blockIdx.x * blockDim.x + threadIdx.x;
  if (e >= EDGES + NODES) return;
  int src, dst;
  edge_fetch(ei, e, src, dst);
  atomicAdd(&s2[dst], __expf(lrelu(as2[src] + ad2[dst]) - m2[dst]));
}

__global__ void k_agg2(const int* __restrict__ ei, const float* __restrict__ as2,
                       const float* __restrict__ ad2, const float* __restrict__ m2,
                       const float* __restrict__ s2, const float* __restrict__ h2,
                       float* __restrict__ acc2) {
  const int e = blockIdx.x * blockDim.x + threadIdx.x;
  if (e >= EDGES + NODES) return;
  int src, dst;
  edge_fetch(ei, e, src, dst);
  const float v = lrelu(as2[src] + ad2[dst]);
  const float alpha = __expf(v - m2[dst]) / (s2[dst] + 1e-16f);
  atomicAdd(&acc2[dst], h2[src] * alpha);
}

__global__ void k_out(const float* __restrict__ acc2, const float* __restrict__ b2,
                      float* __restrict__ out) {
  const int n = blockIdx.x * blockDim.x + threadIdx.x;
  if (n >= NODES) return;
  const float v = acc2[n] + b2[0];
  out[n] = 1.f / (1.f + __expf(-v));
}

// ---------------------------------------------------------------------------
extern "C" void kernel_launch(void* const* d_in, const int* in_sizes, int n_in,
                              void* d_out, int out_size, void* d_ws, size_t ws_size,
                              hipStream_t stream) {
  (void)in_sizes; (void)n_in; (void)out_size; (void)ws_size;

  const float* x        = (const float*)d_in[0];
  const int*   ei       = (const int*)d_in[1];     // [2,E] flat: row0=src, row1=dst
  const float* W1       = (const float*)d_in[2];
  const float* att_src1 = (const float*)d_in[3];
  const float* att_dst1 = (const float*)d_in[4];
  const float* b1       = (const float*)d_in[5];
  const float* W2       = (const float*)d_in[6];
  const float* att_src2 = (const float*)d_in[7];
  const float* att_dst2 = (const float*)d_in[8];
  const float* b2       = (const float*)d_in[9];
  float* out = (float*)d_out;

  float* ws = (float*)d_ws;
  float* h1   = ws; ws += (size_t)NODES * HF;      // 25.6 MB
  float* a_s1 = ws; ws += (size_t)NODES * HEADS;
  float* a_d1 = ws; ws += (size_t)NODES * HEADS;
  float* m1   = ws; ws += (size_t)NODES * HEADS;
  float* s1   = ws; ws += (size_t)NODES * HEADS;
  float* acc1 = ws; ws += (size_t)NODES * HF;      // 25.6 MB
  float* h2   = ws; ws += (size_t)NODES;
  float* as2  = ws; ws += (size_t)NODES;
  float* ad2  = ws; ws += (size_t)NODES;
  float* m2   = ws; ws += (size_t)NODES;
  float* s2   = ws; ws += (size_t)NODES;
  float* acc2 = ws; ws += (size_t)NODES;

  const int ET = EDGES + NODES;
  const int T = 256;

  k_gemm1      <<<ROWS_MAIN / 128, T, 0, stream>>>(x, W1, h1);
  k_gemm1_tail <<<((NODES - ROWS_MAIN) * HF + T - 1) / T, T, 0, stream>>>(x, W1, h1);
  k_att1 <<<(NODES * HEADS + T - 1) / T, T, 0, stream>>>(h1, att_src1, att_dst1, a_s1, a_d1);
  k_init <<<(NODES * HF + T - 1) / T, T, 0, stream>>>(m1, s1, acc1, m2, s2, acc2);

  k_max1 <<<(ET + T - 1) / T, T, 0, stream>>>(ei, a_s1, a_d1, m1);
  k_sum1 <<<(ET + T - 1) / T, T, 0, stream>>>(ei, a_s1, a_d1, m1, s1);
  k_agg1 <<<(ET * HEADS + T - 1) / T, T, 0, stream>>>(ei, a_s1, a_d1, m1, s1, h1, acc1);

  k_fin1 <<<(NODES + T - 1) / T, T, 0, stream>>>(acc1, b1, W2, att_src2, att_dst2, h2, as2, ad2);

  k_max2 <<<(ET + T - 1) / T, T, 0, stream>>>(ei, as2, ad2, m2);
  k_sum2 <<<(ET + T - 1) / T, T, 0, stream>>>(ei, as2, ad2, m2, s2);
  k_agg2 <<<(ET + T - 1) / T, T, 0, stream>>>(ei, as2, ad2, m2, s2, h2, acc2);

  k_out  <<<(NODES + T - 1) / T, T, 0, stream>>>(acc2, b2, out);
}